// forward_warp_64312840291112
// MI455X (gfx1250) — compile-verified
//
#include <hip/hip_runtime.h>

// Problem shape from the reference setup (B,C,H,W fixed by setup_inputs).
constexpr int kB  = 4;
constexpr int kC  = 3;
constexpr int kH  = 1024;
constexpr int kW  = 1920;
constexpr int kHW = kH * kW;

// Tiling: 96x96 source pixels per block, +/-12 px halo accumulated in LDS.
// LDS accumulator: 120*120*3 f32 = 172,800 B (fits the 320 KB/WGP pool).
constexpr int TW  = 96;            // tile width  (1920 % 96 == 0)
constexpr int TH  = 96;            // tile height (11 tiles cover 1024)
constexpr int R   = 12;            // halo radius (2.4 sigma of the flow)
constexpr int LW  = TW + 2 * R;    // 120
constexpr int LH  = TH + 2 * R;    // 120
constexpr int LSZ = LW * LH;       // 14,400 cells per channel
constexpr int NTHREADS = 1024;     // 32 wave32 -> 8 waves per SIMD32
constexpr int PIX_PER_THREAD = (TW * TH) / NTHREADS;  // 9

__global__ __launch_bounds__(NTHREADS) void fwarp_splat_tiled(
    const float*  __restrict__ im0,
    const float2* __restrict__ flow,
    const int*    __restrict__ mode_p,
    float*        __restrict__ out)
{
    __shared__ float acc[kC * LSZ];          // 172,800 B static LDS

    const int mode  = *mode_p;               // uniform scalar load
    const int tid   = threadIdx.x;
    const int tileX = blockIdx.x;
    const int tileY = blockIdx.y;
    const int b     = blockIdx.z;
    const int baseX = tileX * TW - R;        // LDS window origin in image coords
    const int baseY = tileY * TH - R;

    // --- zero the LDS accumulator with wide stores (ds_store_b128)
    {
        float4* acc4 = reinterpret_cast<float4*>(acc);
        constexpr int N4 = (kC * LSZ) / 4;   // 10,800 (kC*LSZ divisible by 4)
        const float4 z = make_float4(0.f, 0.f, 0.f, 0.f);
        for (int i = tid; i < N4; i += NTHREADS) acc4[i] = z;
    }
    __syncthreads();

    const size_t bOff = (size_t)(b * kC) * kHW;
    const float*  imb  = im0  + bOff;
    float*        outb = out  + bOff;
    const float2* flb  = flow + (size_t)b * kHW;

#pragma unroll
    for (int k = 0; k < PIX_PER_THREAD; ++k) {
        const int p  = tid + k * NTHREADS;
        const int py = p / TW;
        const int px = p - py * TW;
        const int gx = tileX * TW + px;
        const int gy = tileY * TH + py;
        if (gy >= kH) continue;              // partial bottom tile row

        const int rem = gy * kW + gx;

        // prefetch next chunk of this thread's inputs (global_prefetch_b8)
        if (k + 1 < PIX_PER_THREAD) {
            const int p2  = p + NTHREADS;
            const int py2 = p2 / TW;
            const int gy2 = tileY * TH + py2;
            if (gy2 < kH) {
                const int rem2 = gy2 * kW + (tileX * TW + (p2 - py2 * TW));
                __builtin_prefetch(&flb[rem2],         0, 0);
                __builtin_prefetch(&imb[rem2],         0, 0);
                __builtin_prefetch(&imb[kHW + rem2],   0, 0);
                __builtin_prefetch(&imb[2*kHW + rem2], 0, 0);
            }
        }

        const float2 f = flb[rem];           // global_load_b64, coalesced
        const float  x = (float)gx + f.x;
        const float  y = (float)gy + f.y;

        const float s0 = imb[rem];
        const float s1 = imb[kHW + rem];
        const float s2 = imb[2 * kHW + rem];

        int   xs[4], ys[4];
        float ws[4];
        int   ncorner;
        if (mode == 1) {                     // nearest: jnp.round -> rintf (RNE)
            xs[0] = (int)rintf(x); ys[0] = (int)rintf(y); ws[0] = 1.0f;
            ncorner = 1;
        } else {                             // bilinear splat
            const float xf = floorf(x), yf = floorf(y);
            const int   x0 = (int)xf,   y0 = (int)yf;
            const float wx1 = x - xf,   wy1 = y - yf;
            const float wx0 = 1.0f - wx1, wy0 = 1.0f - wy1;
            xs[0]=x0;   ys[0]=y0;   ws[0]=wx0*wy0;
            xs[1]=x0+1; ys[1]=y0;   ws[1]=wx1*wy0;
            xs[2]=x0;   ys[2]=y0+1; ws[2]=wx0*wy1;
            xs[3]=x0+1; ys[3]=y0+1; ws[3]=wx1*wy1;
            ncorner = 4;
        }

        for (int c = 0; c < ncorner; ++c) {
            const int   xi = xs[c], yi = ys[c];
            const float wt = ws[c];
            // OOB corners have weight masked to 0 in the reference -> skip.
            if (xi < 0 || xi >= kW || yi < 0 || yi >= kH) continue;

            const int lx = xi - baseX;
            const int ly = yi - baseY;
            if ((unsigned)lx < (unsigned)LW && (unsigned)ly < (unsigned)LH) {
                // common case: privatized accumulation in LDS (ds_add_f32).
                // (+0.0 adds are identity -> no need to test wt here)
                const int o = ly * LW + lx;
                atomicAdd(&acc[o],           s0 * wt);
                atomicAdd(&acc[LSZ + o],     s1 * wt);
                atomicAdd(&acc[2 * LSZ + o], s2 * wt);
            } else if (wt != 0.0f) {
                // rare long-range splat: direct global atomic
                const int o = yi * kW + xi;
                atomicAdd(outb + o,           s0 * wt);
                atomicAdd(outb + kHW + o,     s1 * wt);
                atomicAdd(outb + 2 * kHW + o, s2 * wt);
            }
        }
    }

    __syncthreads();

    // --- flush LDS tile to global with atomics (halo regions overlap between
    // neighboring tiles, so every flush must be an atomic add). Skip exact
    // zeros: out is zero-initialized and +0.0 adds are identity.
    for (int i = tid; i < LSZ; i += NTHREADS) {
        const int ly = i / LW;
        const int lx = i - ly * LW;
        const int gx = baseX + lx;
        const int gy = baseY + ly;
        if ((unsigned)gx < (unsigned)kW && (unsigned)gy < (unsigned)kH) {
            const int o  = gy * kW + gx;
            const float v0 = acc[i];
            const float v1 = acc[LSZ + i];
            const float v2 = acc[2 * LSZ + i];
            if (v0 != 0.0f) atomicAdd(outb + o,           v0);
            if (v1 != 0.0f) atomicAdd(outb + kHW + o,     v1);
            if (v2 != 0.0f) atomicAdd(outb + 2 * kHW + o, v2);
        }
    }
}

extern "C" void kernel_launch(void* const* d_in, const int* in_sizes, int n_in,
                              void* d_out, int out_size, void* d_ws, size_t ws_size,
                              hipStream_t stream) {
    const float*  im0  = (const float*)d_in[0];
    const float2* flow = (const float2*)d_in[1];
    // d_in[2] = flowback: accepted but unused by the forward splat (per reference).
    const int*    mode = (const int*)d_in[3];
    float*        out  = (float*)d_out;

    // Output must start at zero (harness poisons it); memset is capturable and
    // stream-ordered ahead of the atomics.
    hipMemsetAsync(out, 0, (size_t)out_size * sizeof(float), stream);

    dim3 grid(kW / TW,                    // 20
              (kH + TH - 1) / TH,         // 11
              kB);                        // 4  -> 880 blocks
    fwarp_splat_tiled<<<grid, NTHREADS, 0, stream>>>(im0, flow, mode, out);
}